// RadianceNetwork_88373247082700
// MI455X (gfx1250) — compile-verified
//
#include <hip/hip_runtime.h>

typedef float v2f __attribute__((ext_vector_type(2)));
typedef float v8f __attribute__((ext_vector_type(8)));

namespace {
constexpr int kBatch = 64;    // B
constexpr int kC     = 256;   // channels (blocks)
constexpr int kInDim = 135;   // layer-1 K (padded to 136)
constexpr int kHid   = 256;
constexpr int kOut   = 816;
constexpr int kLdx   = 138;   // LDS stride for input tile (>=136, even, bank-friendly)
constexpr int kLdh   = 258;   // LDS stride for hidden activations (even, bank-friendly)
}

// Compute NT 64x16 output column blocks (4 M-tiles x NT N-tiles) sharing A fragments:
//   D[j][mt] += A_lds[mt*16.., K] * W[K, n0 + j*128 ..]
// V_WMMA_F32_16X16X4_F32 fragment layout: lane li(+16*grp) holds
//   A: a.x = A[li][4k+2*grp], a.y = A[li][4k+2*grp+1]
//   B: b.x = W[4k+2*grp][n],  b.y = W[4k+2*grp+1][n],  n = n0+li
//   D: vgpr r -> element [M = r + 8*grp][N = li]
template<int NT>
__device__ __forceinline__ void mma_cols(
    const float* __restrict__ A, int lda,          // LDS activation base / stride
    const float* __restrict__ Wcol, int wstride,   // global W + c*K*N + n0 + li
    int ksteps, int kmax,                          // K/4 steps; clamp for W row index
    int grp, int li,
    v8f acc[NT][4])
{
#pragma unroll 2
    for (int ks = 0; ks < ksteps; ++ks) {
        const int r0 = 4 * ks + 2 * grp;
        const int ra = (r0     < kmax) ? r0       : kmax;
        const int rb = (r0 + 1 < kmax) ? (r0 + 1) : kmax;
        v2f bf[NT];
#pragma unroll
        for (int j = 0; j < NT; ++j) {
            bf[j].x = Wcol[(size_t)ra * wstride + j * 128];
            bf[j].y = Wcol[(size_t)rb * wstride + j * 128];
        }
#pragma unroll
        for (int mt = 0; mt < 4; ++mt) {
            const v2f af = *(const v2f*)(A + (mt * 16 + li) * lda + r0); // ds_load_b64
#pragma unroll
            for (int j = 0; j < NT; ++j)
                acc[j][mt] = __builtin_amdgcn_wmma_f32_16x16x4_f32(
                    false, af, false, bf[j], (short)0, acc[j][mt], false, false);
        }
    }
}

__device__ __forceinline__ void init_bias(const float* __restrict__ b, v8f acc[4]) {
    const float bias = *b;
    const v8f binit = {bias, bias, bias, bias, bias, bias, bias, bias};
#pragma unroll
    for (int mt = 0; mt < 4; ++mt) acc[mt] = binit;
}

__device__ __forceinline__ void store_relu_lds(float* __restrict__ h, int col,
                                               int grp, int li, const v8f acc[4]) {
#pragma unroll
    for (int mt = 0; mt < 4; ++mt)
#pragma unroll
        for (int r = 0; r < 8; ++r) {
            const float v = acc[mt][r];
            h[(mt * 16 + r + 8 * grp) * kLdh + col + li] = (v > 0.0f) ? v : 0.0f;
        }
}

__device__ __forceinline__ void store_out(float* __restrict__ out, int c, int col,
                                          int grp, int li, const v8f acc[4]) {
#pragma unroll
    for (int mt = 0; mt < 4; ++mt)
#pragma unroll
        for (int r = 0; r < 8; ++r) {
            const size_t row = (size_t)(mt * 16 + r + 8 * grp);
            out[row * ((size_t)kC * kOut) + (size_t)c * kOut + col + li] = acc[mt][r];
        }
}

__launch_bounds__(256)
__global__ void radiance_mlp_kernel(
    const float* __restrict__ pos,  const float* __restrict__ view,
    const float* __restrict__ feat, const int*   __restrict__ bsid,
    const float* __restrict__ W1, const float* __restrict__ b1,
    const float* __restrict__ W2, const float* __restrict__ b2,
    const float* __restrict__ W3, const float* __restrict__ b3,
    float* __restrict__ out)
{
    // smA: input x[64 x kLdx] during layer 1, then h2[64 x kLdh] for layers 2->3.
    // smB: h1[64 x kLdh].
    __shared__ float smA[kBatch * kLdh];
    __shared__ float smB[kBatch * kLdh];

    const int c    = blockIdx.x;
    const int tid  = threadIdx.x;
    const int wave = tid >> 5;
    const int lane = tid & 31;
    const int grp  = lane >> 4;   // 0: lanes 0-15, 1: lanes 16-31
    const int li   = lane & 15;

    // ---- Build shared input x[64][138]: [pos(3) | view(3) | feat(128) | nid(1) | 0 pad] ----
    for (int idx = tid; idx < kBatch * kLdx; idx += blockDim.x) {
        const int m = idx / kLdx;
        const int k = idx - m * kLdx;
        float v;
        if (k < 3)         v = pos[m * 3 + k];
        else if (k < 6)    v = view[m * 3 + (k - 3)];
        else if (k < 134)  v = feat[m * 128 + (k - 6)];
        else if (k == 134) v = ((float)bsid[m] - 1.0f) * (1.0f / 63.0f);
        else               v = 0.0f;                    // zero pad cols 135..137
        smA[idx] = v;
    }
    __syncthreads();

    // ---- Layer 1: h1 = relu(x @ W1 + b1), K=136 (clamped), N=256: 2 N-tiles per wave ----
    {
        const int n0 = wave * 16;                       // tiles wave and wave+8 (cols n0, n0+128)
        v8f acc[2][4];
        init_bias(b1 + c * kHid + n0 + li,       acc[0]);
        init_bias(b1 + c * kHid + n0 + 128 + li, acc[1]);
        mma_cols<2>(smA, kLdx, W1 + (size_t)c * kInDim * kHid + n0 + li, kHid,
                    /*ksteps=*/34, /*kmax=*/134, grp, li, acc);
        store_relu_lds(smB, n0,       grp, li, acc[0]);
        store_relu_lds(smB, n0 + 128, grp, li, acc[1]);
    }
    __syncthreads();   // h1 ready; x no longer needed (smA becomes h2)

    // ---- Layer 2: h2 = relu(h1 @ W2 + b2), K=256, N=256: 2 N-tiles per wave ----
    {
        const int n0 = wave * 16;
        v8f acc[2][4];
        init_bias(b2 + c * kHid + n0 + li,       acc[0]);
        init_bias(b2 + c * kHid + n0 + 128 + li, acc[1]);
        mma_cols<2>(smB, kLdh, W2 + (size_t)c * kHid * kHid + n0 + li, kHid,
                    /*ksteps=*/64, /*kmax=*/255, grp, li, acc);
        store_relu_lds(smA, n0,       grp, li, acc[0]);
        store_relu_lds(smA, n0 + 128, grp, li, acc[1]);
    }
    __syncthreads();   // h2 ready

    // ---- Layer 3: out = h2 @ W3 + b3, K=256, N=816 (51 tiles), direct to global ----
    // Passes p=0..2: tile pairs (wave+16p, wave+16p+8) -> 48 tiles. Tail: tiles 48..50.
    for (int p = 0; p < 3; ++p) {
        const int n0 = (wave + 16 * p) * 16;
        v8f acc[2][4];
        init_bias(b3 + c * kOut + n0 + li,       acc[0]);
        init_bias(b3 + c * kOut + n0 + 128 + li, acc[1]);
        mma_cols<2>(smA, kLdh, W3 + (size_t)c * kHid * kOut + n0 + li, kOut,
                    /*ksteps=*/64, /*kmax=*/255, grp, li, acc);
        store_out(out, c, n0,       grp, li, acc[0]);
        store_out(out, c, n0 + 128, grp, li, acc[1]);
    }
    if (wave + 48 < kOut / 16) {                        // waves 0..2: tiles 48..50
        const int n0 = (wave + 48) * 16;
        v8f acc1[1][4];
        init_bias(b3 + c * kOut + n0 + li, acc1[0]);
        mma_cols<1>(smA, kLdh, W3 + (size_t)c * kHid * kOut + n0 + li, kOut,
                    /*ksteps=*/64, /*kmax=*/255, grp, li, acc1);
        store_out(out, c, n0, grp, li, acc1[0]);
    }
}

extern "C" void kernel_launch(void* const* d_in, const int* in_sizes, int n_in,
                              void* d_out, int out_size, void* d_ws, size_t ws_size,
                              hipStream_t stream) {
    const float* pos  = (const float*)d_in[0];
    const float* view = (const float*)d_in[1];
    const float* feat = (const float*)d_in[2];
    const int*   bsid = (const int*)d_in[3];
    const float* W1   = (const float*)d_in[4];
    const float* b1   = (const float*)d_in[5];
    const float* W2   = (const float*)d_in[6];
    const float* b2   = (const float*)d_in[7];
    const float* W3   = (const float*)d_in[8];
    const float* b3   = (const float*)d_in[9];
    float* out = (float*)d_out;

    hipLaunchKernelGGL(radiance_mlp_kernel, dim3(kC), dim3(256), 0, stream,
                       pos, view, feat, bsid, W1, b1, W2, b2, W3, b3, out);
}